// PillarNet_79791902425222
// MI455X (gfx1250) — compile-verified
//
#include <hip/hip_runtime.h>

// PillarNet pillar-feature extraction for MI455X (gfx1250).
// Bandwidth-bound scatter/gather: no matrix math -> no WMMA by design.
// CDNA5 paths used: global_load_async_to_lds_b128 (+ s_wait_asynccnt) for
// staging streaming point data, f32 global atomics into an L2-resident
// accumulator, non-temporal 128-bit output stores.

#define NPTS      4000000
#define NBATCH    16
#define GRID_X    512
#define GRID_Y    512
#define NSEG      (NBATCH * GRID_X * GRID_Y)      // 4,194,304 pillars
#define BLOCK     256
#define PPT       4                               // points per thread (80 B, 16B-aligned)
#define THREADS_TOTAL ((NPTS + PPT - 1) / PPT)    // 1,000,000

typedef __attribute__((ext_vector_type(4))) float v4f;  // native vector for NT builtins

// ---------------------------------------------------------------------------
// Kernel 1: zero the {sum_x, sum_y, sum_z, count} accumulator (64 MiB in d_ws)
// ---------------------------------------------------------------------------
__global__ __launch_bounds__(BLOCK) void pn_zero_ws(v4f* __restrict__ ws)
{
    int i = blockIdx.x * BLOCK + threadIdx.x;
    int stride = gridDim.x * BLOCK;
    const v4f z = {0.f, 0.f, 0.f, 0.f};
    for (; i < NSEG; i += stride)
        ws[i] = z;
}

// ---------------------------------------------------------------------------
// Async-stage this thread's 4 points (5 x 16 B = 80 B) into its LDS slot.
// Per ISA 10.2, the low 32 bits of a flat shared-aperture address are the LDS
// byte offset, which is exactly what the async op's VDST VGPR wants.
// ---------------------------------------------------------------------------
__device__ __forceinline__ void stage_points_async(const float* __restrict__ pts,
                                                   v4f* stage_slot, int g,
                                                   bool active)
{
    unsigned lds = (unsigned)(unsigned long long)(void*)stage_slot;
    if (active) {
        unsigned goff = (unsigned)g * 80u;
#pragma unroll
        for (int k = 0; k < 5; ++k) {
            asm volatile("global_load_async_to_lds_b128 %0, %1, %2"
                         :
                         : "v"(lds + 16u * (unsigned)k),
                           "v"(goff + 16u * (unsigned)k),
                           "s"(pts)
                         : "memory");
        }
    }
    asm volatile("s_wait_asynccnt 0x0" ::: "memory");  // ASYNCcnt == 0
}

// ---------------------------------------------------------------------------
// Decode one point from the staged LDS slot.
// ---------------------------------------------------------------------------
struct PillarPt {
    float x, y, z, r;
    float fx, fy;   // floor(pc_x), floor(pc_y) as floats (unclipped, like ref)
    int   lin;
    bool  m;
};

__device__ __forceinline__ void decode_point(const float* __restrict__ mf, int j,
                                             PillarPt& p)
{
    const float b = mf[5 * j + 0];
    p.x = mf[5 * j + 1];
    p.y = mf[5 * j + 2];
    p.z = mf[5 * j + 3];
    p.r = mf[5 * j + 4];
    // pc = (xyz - pc_range_min) / voxel ; voxel.xy = 0.2 -> * 5.0
    const float pcx = (p.x + 51.2f) * 5.0f;
    const float pcy = (p.y + 51.2f) * 5.0f;
    p.m = (pcx >= 0.f) && (pcx < 512.f) && (pcy >= 0.f) && (pcy < 512.f);
    p.fx = floorf(pcx);
    p.fy = floorf(pcy);
    int ix = (int)p.fx;
    int iy = (int)p.fy;
    ix = min(max(ix, 0), GRID_X - 1);
    iy = min(max(iy, 0), GRID_Y - 1);
    const int bi = (int)b;
    p.lin = p.m ? ((bi * GRID_X + ix) * GRID_Y + iy) : 0;
}

// ---------------------------------------------------------------------------
// Kernel 2: scatter  sum_xyz / count  via f32 global atomics (L2-resident ws)
// ---------------------------------------------------------------------------
__global__ __launch_bounds__(BLOCK) void pn_scatter(const float* __restrict__ pts,
                                                    float* __restrict__ ws)
{
    __shared__ v4f stage[BLOCK * 5];
    const int g = blockIdx.x * BLOCK + threadIdx.x;
    const bool active = g < THREADS_TOTAL;
    stage_points_async(pts, &stage[threadIdx.x * 5], g, active);
    if (!active) return;

    const float* mf = (const float*)&stage[threadIdx.x * 5];
#pragma unroll
    for (int j = 0; j < PPT; ++j) {
        PillarPt p;
        decode_point(mf, j, p);
        if (p.m) {
            float* s = ws + 4 * (size_t)p.lin;
            atomicAdd(s + 0, p.x);     // global_atomic_add_f32 (no return)
            atomicAdd(s + 1, p.y);
            atomicAdd(s + 2, p.z);
            atomicAdd(s + 3, 1.0f);
        }
    }
}

// ---------------------------------------------------------------------------
// Kernel 3: gather mean, emit 9 features/point, NT 128-bit output stores
// ---------------------------------------------------------------------------
__global__ __launch_bounds__(BLOCK) void pn_gather(const float* __restrict__ pts,
                                                   const v4f* __restrict__ ws,
                                                   float* __restrict__ out)
{
    __shared__ v4f stage[BLOCK * 5];
    const int g = blockIdx.x * BLOCK + threadIdx.x;
    const bool active = g < THREADS_TOTAL;
    stage_points_async(pts, &stage[threadIdx.x * 5], g, active);
    if (!active) return;

    const float* mf = (const float*)&stage[threadIdx.x * 5];
    float o[PPT * 9];
#pragma unroll
    for (int j = 0; j < PPT; ++j) {
        PillarPt p;
        decode_point(mf, j, p);
        float* r = &o[9 * j];
        if (p.m) {
            const v4f s = ws[p.lin];                    // global_load_b128 gather
            const float inv = 1.0f / fmaxf(s.w, 1.0f);  // mean = sum / max(cnt,1)
            r[0] = p.x;
            r[1] = p.y;
            r[2] = p.z;
            r[3] = p.r;
            r[4] = p.x - s.x * inv;                     // f_cluster
            r[5] = p.y - s.y * inv;
            r[6] = p.z - s.z * inv;
            r[7] = p.x - (p.fx * 0.2f + (0.1f - 51.2f)); // f_center
            r[8] = p.y - (p.fy * 0.2f + (0.1f - 51.2f));
        } else {
#pragma unroll
            for (int c = 0; c < 9; ++c) r[c] = 0.f;
        }
    }

    // 36 contiguous floats per thread, 144-byte aligned -> 9 NT b128 stores.
    v4f* ob = (v4f*)(out + (size_t)g * (PPT * 9));
    const v4f* ov = (const v4f*)o;
#pragma unroll
    for (int k = 0; k < (PPT * 9) / 4; ++k)
        __builtin_nontemporal_store(ov[k], ob + k);
}

// ---------------------------------------------------------------------------
extern "C" void kernel_launch(void* const* d_in, const int* in_sizes, int n_in,
                              void* d_out, int out_size, void* d_ws, size_t ws_size,
                              hipStream_t stream)
{
    (void)in_sizes; (void)n_in; (void)out_size; (void)ws_size;
    const float* pts = (const float*)d_in[0];
    float*       out = (float*)d_out;

    // Workspace: NSEG float4 accumulators = 64 MiB (fits in 192 MB L2).
    pn_zero_ws<<<4096, BLOCK, 0, stream>>>((v4f*)d_ws);

    const int nb = (THREADS_TOTAL + BLOCK - 1) / BLOCK;   // 3907 blocks
    pn_scatter<<<nb, BLOCK, 0, stream>>>(pts, (float*)d_ws);
    pn_gather<<<nb, BLOCK, 0, stream>>>(pts, (const v4f*)d_ws, out);
}